// GraphCluster_25305947308740
// MI455X (gfx1250) — compile-verified
//
#include <hip/hip_runtime.h>
#include <hip/hip_bf16.h>

#define N_FEAT 128

typedef float v2f __attribute__((ext_vector_type(2)));
typedef float v8f __attribute__((ext_vector_type(8)));

// ---------------------------------------------------------------------------
// Degree / normalization kernels
// ---------------------------------------------------------------------------
__global__ void deg_init_k(float* __restrict__ deg, int n) {
    int i = blockIdx.x * 256 + threadIdx.x;
    if (i < n) deg[i] = 1.0f;  // self-loop contribution
}

__global__ void deg_count_k(const int* __restrict__ dst, float* __restrict__ deg, int e) {
    int i = blockIdx.x * 256 + threadIdx.x;
    if (i < e) atomicAdd(&deg[dst[i]], 1.0f);
}

__global__ void dinv_fin_k(float* __restrict__ deg, int n) {
    int i = blockIdx.x * 256 + threadIdx.x;
    if (i < n) deg[i] = rsqrtf(fmaxf(deg[i], 1.0f));
}

// ---------------------------------------------------------------------------
// WMMA fp32 GEMM:  out[M x NCOLS] = A[M x 128] @ W[128 x NCOLS]   (K = 128)
// MODE 0: plain            MODE 1: sigmoid(x + bias)
// Register blocking: each wave owns MT=2 m-tiles x NT n-tiles.
// NCOLS is compile-time -> all DS/global offsets become immediates.
// ---------------------------------------------------------------------------
template <int NCOLS, int MODE>
__global__ __launch_bounds__(256) void gemm_wmma_k(
    const float* __restrict__ A, const float* __restrict__ W,
    const float* __restrict__ bias, float* __restrict__ out, int M)
{
    constexpr int MT = 2;
    constexpr int NT = (NCOLS >= 64) ? 4 : 1;
    constexpr int NPASS = NCOLS / (16 * NT);

    __shared__ float sW[N_FEAT * NCOLS];  // 64 KB (NCOLS=128) / 8 KB (NCOLS=16)

    const int t = threadIdx.x;
    #pragma unroll 4
    for (int i = t; i < N_FEAT * NCOLS; i += 256) sW[i] = W[i];
    __syncthreads();

    const int wave  = t >> 5;
    const int lane  = t & 31;
    const int row16 = lane & 15;   // A row / B,C,D column within tile
    const int khalf = lane >> 4;   // 0: K pair {0,1}; 1: K pair {2,3}

    const int mtile0 = (blockIdx.x * 8 + wave) * MT;
    if (mtile0 * 16 >= M) return;

    // A row pointers per m-tile (clamped; M=100000 is an exact tile multiple)
    const float* arow[MT];
    #pragma unroll
    for (int mi = 0; mi < MT; ++mi) {
        int r = (mtile0 + mi) * 16 + row16;
        if (r > M - 1) r = M - 1;
        arow[mi] = A + (size_t)r * N_FEAT + 2 * khalf;
    }

    #pragma unroll
    for (int pass = 0; pass < NPASS; ++pass) {
        const int n0p = pass * NT * 16;

        v8f acc[MT][NT];
        #pragma unroll
        for (int mi = 0; mi < MT; ++mi)
            #pragma unroll
            for (int ni = 0; ni < NT; ++ni)
                acc[mi][ni] = (v8f){};

        #pragma unroll 8
        for (int k0 = 0; k0 < N_FEAT; k0 += 4) {
            // A fragments: lane<16 -> K={k0,k0+1}, lane>=16 -> K={k0+2,k0+3}
            v2f a[MT];
            #pragma unroll
            for (int mi = 0; mi < MT; ++mi)
                a[mi] = *(const v2f*)(arow[mi] + k0);

            #pragma unroll
            for (int ni = 0; ni < NT; ++ni) {
                const int col = n0p + ni * 16 + row16;
                // B fragment: v0 = rows {k0, k0+2}, v1 = rows {k0+1, k0+3}
                v2f b;
                b.x = sW[(k0 + 2 * khalf) * NCOLS + col];
                b.y = sW[(k0 + 2 * khalf + 1) * NCOLS + col];
                #pragma unroll
                for (int mi = 0; mi < MT; ++mi)
                    acc[mi][ni] = __builtin_amdgcn_wmma_f32_16x16x4_f32(
                        false, a[mi], false, b, (short)0, acc[mi][ni],
                        false, false);
            }
        }

        // Epilogue. D layout: VGPR j -> row j (+8 for hi lanes), col = lane%16
        #pragma unroll
        for (int ni = 0; ni < NT; ++ni) {
            const int col = n0p + ni * 16 + row16;
            const float bval = (MODE == 1) ? bias[col] : 0.0f;
            #pragma unroll
            for (int mi = 0; mi < MT; ++mi) {
                const int rbase = (mtile0 + mi) * 16 + 8 * khalf;
                #pragma unroll
                for (int j = 0; j < 8; ++j) {
                    const int r = rbase + j;
                    if (r < M) {
                        float v = acc[mi][ni][j];
                        if (MODE == 1) {
                            v += bval;
                            v = 1.0f / (1.0f + __expf(-v));
                        }
                        out[(size_t)r * NCOLS + col] = v;
                    }
                }
            }
        }
    }
}

// ---------------------------------------------------------------------------
// out[i,f] = xw[i,f] * dinv[i]^2 + bias[f]   (self-loop term + bias, full init)
// ---------------------------------------------------------------------------
__global__ void selfloop_init_k(const float* __restrict__ xw,
                                const float* __restrict__ dinv,
                                const float* __restrict__ bias,
                                float* __restrict__ out, int n, int F)
{
    size_t i = (size_t)blockIdx.x * 256 + threadIdx.x;
    size_t tot = (size_t)n * F;
    if (i < tot) {
        int node = (int)(i / F);
        int f    = (int)(i % F);
        float di = dinv[node];
        out[i] = xw[i] * di * di + bias[f];
    }
}

// ---------------------------------------------------------------------------
// Edge scatter, F=128: one wave per edge, float4 per lane
// out[dst] += xw[src] * dinv[src]*dinv[dst]
// ---------------------------------------------------------------------------
__global__ void edge_scatter128_k(const int* __restrict__ src,
                                  const int* __restrict__ dst,
                                  const float* __restrict__ dinv,
                                  const float* __restrict__ xw,
                                  float* __restrict__ out, int e)
{
    int g = blockIdx.x * 256 + threadIdx.x;
    int edge = g >> 5;
    int lane = g & 31;
    if (edge >= e) return;
    int s = src[edge], d = dst[edge];
    float w = dinv[s] * dinv[d];
    const float4 v = ((const float4*)(xw + (size_t)s * N_FEAT))[lane];
    float* o = out + (size_t)d * N_FEAT + lane * 4;
    atomicAdd(o + 0, v.x * w);
    atomicAdd(o + 1, v.y * w);
    atomicAdd(o + 2, v.z * w);
    atomicAdd(o + 3, v.w * w);
}

// ---------------------------------------------------------------------------
// Edge scatter, F=16: one thread per (edge, feature)
// ---------------------------------------------------------------------------
__global__ void edge_scatter16_k(const int* __restrict__ src,
                                 const int* __restrict__ dst,
                                 const float* __restrict__ dinv,
                                 const float* __restrict__ xw,
                                 float* __restrict__ out, int e)
{
    int g = blockIdx.x * 256 + threadIdx.x;
    int edge = g >> 4;
    int f    = g & 15;
    if (edge >= e) return;
    int s = src[edge], d = dst[edge];
    float w = dinv[s] * dinv[d];
    atomicAdd(&out[(size_t)d * 16 + f], xw[(size_t)s * 16 + f] * w);
}

// ---------------------------------------------------------------------------
// host launcher
// ---------------------------------------------------------------------------
extern "C" void kernel_launch(void* const* d_in, const int* in_sizes, int n_in,
                              void* d_out, int out_size, void* d_ws, size_t ws_size,
                              hipStream_t stream)
{
    const int*   adj   = (const int*)  d_in[0];
    const float* X     = (const float*)d_in[1];
    const float* fc1_W = (const float*)d_in[2];
    const float* fc1_b = (const float*)d_in[3];
    const float* fc2_W = (const float*)d_in[4];
    const float* fc2_b = (const float*)d_in[5];
    const float* gcn_W = (const float*)d_in[6];
    const float* gcn_b = (const float*)d_in[7];
    const float* asn_W = (const float*)d_in[8];
    const float* asn_b = (const float*)d_in[9];
    float* out = (float*)d_out;

    const int E = in_sizes[0] / 2;
    const int N = in_sizes[1] / N_FEAT;

    const int* src = adj;        // adj[0]
    const int* dst = adj + E;    // adj[1]

    // workspace: two N x 128 ping-pong buffers + dinv[N]
    float* bufA = (float*)d_ws;
    float* bufB = bufA + (size_t)N * N_FEAT;
    float* dinv = bufB + (size_t)N * N_FEAT;

    const int nB = (N + 255) / 256;
    const int eB = (E + 255) / 256;

    // ---- symmetric normalization: dinv = rsqrt(1 + indegree) ----
    deg_init_k <<<nB, 256, 0, stream>>>(dinv, N);
    deg_count_k<<<eB, 256, 0, stream>>>(dst, dinv, E);
    dinv_fin_k <<<nB, 256, 0, stream>>>(dinv, N);

    const int mtiles  = (N + 15) / 16;
    const int gblocks = (mtiles + 16 - 1) / 16;  // 8 waves x MT=2 tiles per block

    // ---- MLP encoder (fused bias + sigmoid) ----
    gemm_wmma_k<N_FEAT, 1><<<gblocks, 256, 0, stream>>>(X,    fc1_W, fc1_b, bufA, N);
    gemm_wmma_k<N_FEAT, 1><<<gblocks, 256, 0, stream>>>(bufA, fc2_W, fc2_b, bufB, N);
    // h now lives in bufB

    // ---- 3 GCN layers: h in bufB -> xw in bufA -> aggregate back into bufB ----
    const size_t nf128 = (size_t)N * N_FEAT;
    const int initB128 = (int)((nf128 + 255) / 256);
    const int scatB128 = (int)(((size_t)E * 32 + 255) / 256);
    for (int l = 0; l < 3; ++l) {
        gemm_wmma_k<N_FEAT, 0><<<gblocks, 256, 0, stream>>>(
            bufB, gcn_W + (size_t)l * N_FEAT * N_FEAT, nullptr, bufA, N);
        selfloop_init_k<<<initB128, 256, 0, stream>>>(
            bufA, dinv, gcn_b + (size_t)l * N_FEAT, bufB, N, N_FEAT);
        edge_scatter128_k<<<scatB128, 256, 0, stream>>>(src, dst, dinv, bufA, bufB, E);
    }

    // ---- assignment layer (N x 16) straight into d_out ----
    gemm_wmma_k<16, 0><<<gblocks, 256, 0, stream>>>(bufB, asn_W, nullptr, bufA, N);
    const size_t nf16 = (size_t)N * 16;
    selfloop_init_k<<<(int)((nf16 + 255) / 256), 256, 0, stream>>>(bufA, dinv, asn_b, out, N, 16);
    edge_scatter16_k<<<(int)(((size_t)E * 16 + 255) / 256), 256, 0, stream>>>(src, dst, dinv, bufA, out, E);
}